// FeatureConsistencyLoss_13194139533364
// MI455X (gfx1250) — compile-verified
//
#include <hip/hip_runtime.h>
#include <hip/hip_bf16.h>

#define B_    2
#define T_    8
#define C_    64
#define H_    128
#define W_    160
#define H0_   256
#define W0_   320
#define HW_   (H_ * W_)        // 20480
#define NPTS_ (T_ * HW_)       // 163840 points per batch
#define KEYS_ (1 << 21)        // 7-bit voxel coords packed x<<14|y<<7|z
#define EPSF  1e-6f

typedef float v2f __attribute__((ext_vector_type(2)));
typedef float v8f __attribute__((ext_vector_type(8)));

__device__ inline bool finite_f(float a) { return fabsf(a) <= 3.0e38f; } // false for NaN/Inf

// jax.image.resize(method="linear"), antialiased, half-pixel, scale=0.5 exactly:
// separable 4-tap triangle filter, taps {-1,0,1,2} around 2*dst, weights {.25,.75,.75,.25},
// renormalized at edges.
__device__ inline bool resize_pt(const float* __restrict__ pts, int b, int t, int h, int w,
                                 float q[3]) {
  const int   offs[4] = {-1, 0, 1, 2};
  const float wts[4]  = {0.25f, 0.75f, 0.75f, 0.25f};
  float a0 = 0.f, a1 = 0.f, a2 = 0.f, wsum = 0.f;
  int ybase = 2 * h, xbase = 2 * w;
  size_t img = (size_t)(b * T_ + t) * H0_ * W0_;
#pragma unroll
  for (int iy = 0; iy < 4; ++iy) {
    int y = ybase + offs[iy];
    if (y < 0 || y >= H0_) continue;
#pragma unroll
    for (int ix = 0; ix < 4; ++ix) {
      int x = xbase + offs[ix];
      if (x < 0 || x >= W0_) continue;
      float wgt = wts[iy] * wts[ix];
      const float* p = pts + (img + (size_t)y * W0_ + x) * 3;
      a0 += wgt * p[0]; a1 += wgt * p[1]; a2 += wgt * p[2];
      wsum += wgt;
    }
  }
  float inv = 1.0f / wsum;
  q[0] = a0 * inv; q[1] = a1 * inv; q[2] = a2 * inv;
  return finite_f(q[0]) && finite_f(q[1]) && finite_f(q[2]);
}

__device__ inline bool point_key(const float* __restrict__ pts, const int* __restrict__ vmask,
                                 int b, int p, int& key, int& t, int& hw) {
  t = p / HW_; hw = p - t * HW_;
  int h = hw / W_, w = hw - h * W_;
  int m = vmask[((size_t)(b * T_ + t) * H0_ + 2 * h) * W0_ + 2 * w];  // nearest: floor(dst*2)
  float q[3];
  bool fin = resize_pt(pts, b, t, h, w, q);
  if (!m || !fin) return false;
  int vx = min(127, max(0, (int)roundf(q[0] * 20.0f)));  // 1/VOXEL_SIZE = 20
  int vy = min(127, max(0, (int)roundf(q[1] * 20.0f)));
  int vz = min(127, max(0, (int)roundf(q[2] * 20.0f)));
  key = (vx << 14) | (vy << 7) | vz;
  return true;
}

// Pass 1: per-point channel L2 inverse norm (clamped at 1e-12).
__global__ void k_invnorm(const float* __restrict__ feat, float* __restrict__ invn) {
  int p = blockIdx.x * blockDim.x + threadIdx.x;   // 0 .. B*NPTS-1 (exact grid)
  int b = p / NPTS_; int pp = p - b * NPTS_;
  int t = pp / HW_;  int hw = pp - t * HW_;
  const float* f = feat + (size_t)(b * T_ + t) * C_ * HW_ + hw;
  float s = 0.f;
#pragma unroll
  for (int c = 0; c < C_; ++c) { float v = f[(size_t)c * HW_]; s += v * v; }
  invn[p] = 1.0f / fmaxf(sqrtf(s), 1e-12f);
}

// Pass 2a: mark occupied voxel keys among valid points; count valid points.
__global__ void k_mark(const float* __restrict__ pts, const int* __restrict__ vmask,
                       int* __restrict__ gid_table, int* __restrict__ validcnt, int b) {
  int p = blockIdx.x * blockDim.x + threadIdx.x;
  int key, t, hw;
  if (point_key(pts, vmask, b, p, key, t, hw)) {
    gid_table[key] = 1;            // benign race: all write 1 (table preset to -1)
    atomicAdd(validcnt, 1);
  }
}

// Pass 2b: allocate compact group ids.
__global__ void k_alloc(int* __restrict__ gid_table, int* __restrict__ ngroups) {
  int k = blockIdx.x * blockDim.x + threadIdx.x;
  if (gid_table[k] == 1) gid_table[k] = atomicAdd(ngroups, 1);
}

// Pass 2c: accumulate per-group count, view bitmask, conf sum, conf-weighted feature sum.
__global__ void k_accum(const float* __restrict__ feat, const float* __restrict__ pts,
                        const float* __restrict__ conf, const int* __restrict__ vmask,
                        const float* __restrict__ invn, const int* __restrict__ gid_table,
                        int* __restrict__ gcount, unsigned* __restrict__ gview,
                        float* __restrict__ gden, float* __restrict__ gpsum, int b) {
  int p = blockIdx.x * blockDim.x + threadIdx.x;
  int key, t, hw;
  if (!point_key(pts, vmask, b, p, key, t, hw)) return;
  int gid = gid_table[key];
  atomicAdd(&gcount[gid], 1);
  atomicOr(&gview[gid], 1u << t);
  float cv = fmaxf(conf[(size_t)(b * T_ + t) * HW_ + hw], EPSF);
  atomicAdd(&gden[gid], cv);
  float inv = invn[b * NPTS_ + p];
  const float* f = feat + (size_t)(b * T_ + t) * C_ * HW_ + hw;
  float* ps = gpsum + (size_t)gid * C_;
#pragma unroll 8
  for (int c = 0; c < C_; ++c) atomicAdd(&ps[c], f[(size_t)c * HW_] * inv * cv);
}

// Pass 2d: per-group validity flag + normalized prototype (in place).
__global__ void k_stats(int* __restrict__ gcount, const unsigned* __restrict__ gview,
                        const float* __restrict__ gden, float* __restrict__ gpsum,
                        const int* __restrict__ ngroups, float* __restrict__ nvg) {
  int g = blockIdx.x * blockDim.x + threadIdx.x;
  if (g >= *ngroups) { gcount[g] = 0; return; }
  int flag = (gcount[g] >= 2) && (__popc(gview[g]) >= 2);
  float den = fmaxf(gden[g], EPSF);
  float* ps = gpsum + (size_t)g * C_;
  float s = 0.f;
#pragma unroll
  for (int c = 0; c < C_; ++c) { float u = ps[c] / den; s += u * u; }
  float inv = 1.0f / fmaxf(sqrtf(s), 1e-12f);
#pragma unroll
  for (int c = 0; c < C_; ++c) ps[c] = (ps[c] / den) * inv;
  gcount[g] = flag;                 // reuse as selection flag for pass 3
  if (flag) atomicAdd(nvg, 1.0f);
}

__device__ inline float v8_get(v8f v, int i) {
  switch (i & 7) {
    case 0: return v[0]; case 1: return v[1]; case 2: return v[2]; case 3: return v[3];
    case 4: return v[4]; case 5: return v[5]; case 6: return v[6]; default: return v[7];
  }
}

// Pass 3: per-point loss via fp32 WMMA. One wave = 16 points; LDS-staged
// F(16x64) x P^T(64x16) with 16 chained V_WMMA_F32_16X16X4_F32; diagonal = dots.
__global__ __launch_bounds__(32)
void k_dot_wmma(const float* __restrict__ feat, const float* __restrict__ pts,
                const int* __restrict__ vmask, const float* __restrict__ invn,
                const int* __restrict__ gid_table, const int* __restrict__ gflag,
                const float* __restrict__ proto, float* __restrict__ ml,
                float* __restrict__ nsel, int b) {
  __shared__ float sF[16][C_];
  __shared__ float sP[16][C_];
  __shared__ int   sSel[16];
  __shared__ int   sT[16];
  __shared__ int   sHW[16];
  __shared__ int   sGid[16];
  int lane = threadIdx.x;
  int p0 = blockIdx.x * 16;

  if (lane < 16) {
    int key, t, hw;
    bool valid = point_key(pts, vmask, b, p0 + lane, key, t, hw);
    int gid = -1, sel = 0;
    if (valid) { gid = gid_table[key]; sel = (gid >= 0) && gflag[gid]; }
    sT[lane] = t; sHW[lane] = hw; sGid[lane] = gid; sSel[lane] = sel;
  }
  __syncthreads();

  for (int idx = lane; idx < 16 * C_; idx += 32) {
    int i = idx >> 6, c = idx & 63;
    float fv = feat[(size_t)(b * T_ + sT[i]) * C_ * HW_ + (size_t)c * HW_ + sHW[i]]
             * invn[b * NPTS_ + p0 + i];
    sF[i][c] = fv;
    sP[i][c] = sSel[i] ? proto[(size_t)sGid[i] * C_ + c] : 0.0f;
  }
  __syncthreads();

  // A (16x4 f32): lanes 0-15 -> K=0,1 ; lanes 16-31 -> K=2,3. B (4x16) mirrored.
  int r = lane & 15;
  int koff = (lane >> 4) * 2;
  v8f acc = {0.f, 0.f, 0.f, 0.f, 0.f, 0.f, 0.f, 0.f};
#pragma unroll
  for (int kb = 0; kb < C_; kb += 4) {
    v2f av, bv;
    av[0] = sF[r][kb + koff];     av[1] = sF[r][kb + koff + 1];
    bv[0] = sP[r][kb + koff];     bv[1] = sP[r][kb + koff + 1];
    acc = __builtin_amdgcn_wmma_f32_16x16x4_f32(false, av, false, bv,
                                                (short)0, acc, false, false);
  }

  // Diagonal (M=N=i): i<8 -> VGPR i, lane i ; i>=8 -> VGPR i-8, lane i+16.
  int pi = -1; float d = 0.0f;
  if (lane < 8)        { pi = lane;       d = v8_get(acc, lane);      }
  else if (lane >= 24) { pi = lane - 16;  d = v8_get(acc, lane - 24); }
  if (pi >= 0 && sSel[pi]) {
    atomicAdd(ml, 1.0f - d);
    atomicAdd(nsel, 1.0f);
  }
}

__global__ void k_final(const float* __restrict__ ml, const float* __restrict__ nsel,
                        const float* __restrict__ nvg, const int* __restrict__ validcnt,
                        const int* __restrict__ ngroups, float* __restrict__ out) {
  float tl = 0.f, tg = 0.f, tvg = 0.f, ts = 0.f;
  for (int b = 0; b < B_; ++b) {
    if (validcnt[b] >= 2) {   // torch skips batch unless >= MIN_GROUP_SIZE valid points
      tl += ml[b]; tg += (float)ngroups[b]; tvg += nvg[b]; ts += nsel[b];
    }
  }
  out[0] = (ts > 0.0f) ? tl / fmaxf(ts, 1.0f) : 0.0f;
  out[1] = tg; out[2] = tvg; out[3] = ts;
}

extern "C" void kernel_launch(void* const* d_in, const int* in_sizes, int n_in,
                              void* d_out, int out_size, void* d_ws, size_t ws_size,
                              hipStream_t stream) {
  (void)in_sizes; (void)n_in; (void)out_size; (void)ws_size;
  const float* feat  = (const float*)d_in[0];
  const float* pts   = (const float*)d_in[1];
  const float* conf  = (const float*)d_in[2];
  const int*   vmask = (const int*)d_in[3];
  float* out = (float*)d_out;
  char*  ws  = (char*)d_ws;

  // Workspace layout (~53.6 MB total; L2-resident on MI455X's 192 MB L2):
  float* ml       = (float*)ws;              // [2]
  float* nsel     = ml + 2;                  // [2]
  float* nvg      = nsel + 2;                // [2]
  int*   validcnt = (int*)(nvg + 2);         // [2]
  int*   ngroups  = validcnt + 2;            // [2]
  float* invn      = (float*)(ws + 256);                         // B*NPTS
  int*   gid_table = (int*)(ws + 256 + (size_t)B_ * NPTS_ * 4);  // KEYS (reused per batch)
  int*       gcount = gid_table + KEYS_;                         // NPTS
  unsigned*  gview  = (unsigned*)(gcount + NPTS_);               // NPTS
  float*     gden   = (float*)(gview + NPTS_);                   // NPTS
  float*     gpsum  = gden + NPTS_;                              // NPTS*64

  hipMemsetAsync(ws, 0, 256, stream);
  k_invnorm<<<(B_ * NPTS_) / 256, 256, 0, stream>>>(feat, invn);

  for (int b = 0; b < B_; ++b) {   // stream order serializes -> workspace reuse
    hipMemsetAsync(gid_table, 0xFF, (size_t)KEYS_ * 4, stream);
    hipMemsetAsync(gcount, 0, (size_t)NPTS_ * 4 * 3 + (size_t)NPTS_ * C_ * 4, stream);
    k_mark <<<NPTS_ / 256, 256, 0, stream>>>(pts, vmask, gid_table, validcnt + b, b);
    k_alloc<<<KEYS_ / 256, 256, 0, stream>>>(gid_table, ngroups + b);
    k_accum<<<NPTS_ / 256, 256, 0, stream>>>(feat, pts, conf, vmask, invn, gid_table,
                                             gcount, gview, gden, gpsum, b);
    k_stats<<<NPTS_ / 256, 256, 0, stream>>>(gcount, gview, gden, gpsum, ngroups + b, nvg + b);
    k_dot_wmma<<<NPTS_ / 16, 32, 0, stream>>>(feat, pts, vmask, invn, gid_table,
                                              gcount, gpsum, ml + b, nsel + b, b);
  }
  k_final<<<1, 1, 0, stream>>>(ml, nsel, nvg, validcnt, ngroups, out);
}